// GCNConv_54348516163921
// MI455X (gfx1250) — compile-verified
//
#include <hip/hip_runtime.h>

typedef __attribute__((ext_vector_type(2))) float        v2f;
typedef __attribute__((ext_vector_type(4))) float        v4f;
typedef __attribute__((ext_vector_type(8))) float        v8f;
typedef __attribute__((ext_vector_type(4))) unsigned int v4u;
typedef __attribute__((ext_vector_type(4))) int          v4i;
typedef __attribute__((ext_vector_type(8))) int          v8i;

#define N_NODES 100000
#define N_EDGES 3200000
#define D_IN    256
#define D_OUT   256
#define WTILE   64          // columns of W staged per workgroup

// ---------------------------------------------------------------------------
// Kernel 0: out[r][c] = b[c]   (atomics accumulate on top of the bias)
// ---------------------------------------------------------------------------
__global__ __launch_bounds__(256) void gcn_init_out(const float* __restrict__ b,
                                                    float* __restrict__ out) {
    int idx = blockIdx.x * 256 + threadIdx.x;
    if (idx < N_NODES * D_OUT) {
        out[idx] = b[idx & (D_OUT - 1)];
    }
}

// ---------------------------------------------------------------------------
// Kernel 1: H = X @ W  via V_WMMA_F32_16X16X4_F32 (exact fp32 matrix pipe).
//
// Per workgroup: TDM (tensor_load_to_lds) stages the 256x64 slab of W used by
// this block into LDS (64 KB) — zero vmem/VALU cost, tracked on TENSORcnt.
// Per wave: 16-row x 64-col output slab, K-loop of wmma_f32_16x16x4 with
// A-fragments streamed from global (HBM) and B-fragments read from LDS.
//
// A-fragment (16x4 f32, 2 VGPRs/lane): lanes 0-15 hold K=kk+0/1 for row=lane,
// lanes 16-31 hold K=kk+2/3 for row=lane-16. B mirrors. C/D: VGPR g = row g
// (lanes 0-15) / row g+8 (lanes 16-31), col = lane&15.
// ---------------------------------------------------------------------------
__global__ __launch_bounds__(256) void gcn_gemm_wmma(const float* __restrict__ X,
                                                     const float* __restrict__ W,
                                                     float* __restrict__ H) {
    __shared__ float lds_w[D_IN * WTILE];   // [k][0..63], row-major, 64 KB

    const int lane  = threadIdx.x & 31;
    const int wave  = threadIdx.x >> 5;
    const int m16   = lane & 15;
    const int khalf = lane >> 4;            // 0 for lanes 0-15, 1 for lanes 16-31

    const int col0 = blockIdx.y * WTILE;    // this block's 64-column slab of W

    // ---- Stage W[:, col0:col0+64] into LDS with the Tensor Data Mover ----
    if (wave == 0) {
        const unsigned int       lds_off = (unsigned int)(size_t)(void*)lds_w;
        const unsigned long long gaddr   = (unsigned long long)(const void*)(W + col0);

        v4u g0;
        g0.x = 0x1u;                                     // count=1 (valid user D#)
        g0.y = lds_off;                                  // lds_addr (bytes)
        g0.z = (unsigned int)gaddr;                      // global_addr[31:0]
        g0.w = (unsigned int)(gaddr >> 32) | (2u << 30); // global_addr[56:32] | type=2

        v8i g1;
        g1[0] = (int)(2u << 16);              // workgroup_mask=0 | data_size=2 (4 bytes)
        g1[1] = (int)(256u << 16);            // tensor_dim0 = 256 (bits 79:48, low half)
        g1[2] = (int)(256u << 16);            // tensor_dim0 hi=0 | tensor_dim1 = 256 (low)
        g1[3] = (int)((unsigned)WTILE << 16); // tensor_dim1 hi=0 | tile_dim0 = 64
        g1[4] = 256;                          // tile_dim1 = 256, tile_dim2 = 0
        g1[5] = 256;                          // tensor_dim0_stride = 256 (low 32)
        g1[6] = 0;                            // stride hi | tensor_dim1_stride (unused)
        g1[7] = 0;

        v4i gz4 = {0, 0, 0, 0};               // groups 2/3 unused (<=2D tensor)
        v8i gz8 = {0, 0, 0, 0, 0, 0, 0, 0};   // extra group (clang-23 6-arg form)

        __builtin_amdgcn_tensor_load_to_lds(g0, g1, gz4, gz4, gz8, 0);
        __builtin_amdgcn_s_wait_tensorcnt(0);
    }
    __syncthreads();                    // LDS now holds the W slab for all waves

    const int rowTile = blockIdx.x * 8 + wave;      // 16-row tile of X
    if (rowTile * 16 >= N_NODES) return;            // wave-uniform guard
    const int row0 = rowTile * 16;

    v8f acc0 = {}, acc1 = {}, acc2 = {}, acc3 = {};

    const float* aPtr = X + (size_t)(row0 + m16) * D_IN + 2 * khalf;
    const float* wl0  = lds_w + (size_t)(2 * khalf) * WTILE + m16;

    #pragma unroll 4
    for (int kk = 0; kk < D_IN; kk += 4) {
        v2f a = *(const v2f*)(aPtr + kk);

        const float* wl = wl0 + (size_t)kk * WTILE;   // row kk+2*khalf of slab
        v2f b0, b1, b2, b3;
        b0.x = wl[0];  b0.y = wl[WTILE + 0];
        b1.x = wl[16]; b1.y = wl[WTILE + 16];
        b2.x = wl[32]; b2.y = wl[WTILE + 32];
        b3.x = wl[48]; b3.y = wl[WTILE + 48];

        acc0 = __builtin_amdgcn_wmma_f32_16x16x4_f32(false, a, false, b0, (short)0, acc0, false, false);
        acc1 = __builtin_amdgcn_wmma_f32_16x16x4_f32(false, a, false, b1, (short)0, acc1, false, false);
        acc2 = __builtin_amdgcn_wmma_f32_16x16x4_f32(false, a, false, b2, (short)0, acc2, false, false);
        acc3 = __builtin_amdgcn_wmma_f32_16x16x4_f32(false, a, false, b3, (short)0, acc3, false, false);
    }

    // Store: VGPR g of each accumulator is row (row0 + 8*khalf + g)
    float* hBase = H + (size_t)(row0 + 8 * khalf) * D_OUT + col0 + m16;
    #pragma unroll
    for (int g = 0; g < 8; ++g) {
        float* hp = hBase + (size_t)g * D_OUT;
        hp[0]  = acc0[g];
        hp[16] = acc1[g];
        hp[32] = acc2[g];
        hp[48] = acc3[g];
    }
}

// ---------------------------------------------------------------------------
// Kernel 2: out[row[e]] += val[e] * H[col[e]]   (H hot in 192MB L2)
// One wave per edge; lanes sweep 128 contiguous floats per b128 load so the
// gather is perfectly coalesced; scatter is global_atomic_add_f32 into L2.
// ---------------------------------------------------------------------------
__global__ __launch_bounds__(256) void gcn_spmm_scatter(const int* __restrict__ erow,
                                                        const int* __restrict__ ecol,
                                                        const float* __restrict__ eval,
                                                        const float* __restrict__ H,
                                                        float* __restrict__ out) {
    const int lane = threadIdx.x & 31;
    const int wave = threadIdx.x >> 5;
    const long e = (long)blockIdx.x * 8 + wave;
    if (e >= N_EDGES) return;

    const int   r = erow[e];
    const int   c = ecol[e];
    const float v = eval[e];

    const float* hp = H   + (size_t)c * D_OUT;
    float*       op = out + (size_t)r * D_OUT;

    #pragma unroll
    for (int j = 0; j < 2; ++j) {
        const int ch = j * 128 + lane * 4;
        v4f hv = *(const v4f*)(hp + ch);
        atomicAdd(op + ch + 0, v * hv.x);
        atomicAdd(op + ch + 1, v * hv.y);
        atomicAdd(op + ch + 2, v * hv.z);
        atomicAdd(op + ch + 3, v * hv.w);
    }
}

// ---------------------------------------------------------------------------
extern "C" void kernel_launch(void* const* d_in, const int* in_sizes, int n_in,
                              void* d_out, int out_size, void* d_ws, size_t ws_size,
                              hipStream_t stream) {
    const float* x        = (const float*)d_in[0];   // [N_NODES, D_IN]
    const int*   edge_row = (const int*)  d_in[1];   // [N_EDGES]
    const int*   edge_col = (const int*)  d_in[2];   // [N_EDGES]
    const float* edge_val = (const float*)d_in[3];   // [N_EDGES]
    const float* weight   = (const float*)d_in[4];   // [D_IN, D_OUT]
    const float* bias     = (const float*)d_in[5];   // [D_OUT]
    float*       out      = (float*)d_out;           // [N_NODES, D_OUT]

    float* h = (float*)d_ws;                         // [N_NODES, D_OUT] = 100 MB scratch

    // 0) out = broadcast(b)
    {
        int total = N_NODES * D_OUT;
        gcn_init_out<<<(total + 255) / 256, 256, 0, stream>>>(bias, out);
    }

    // 1) h = x @ W  (TDM-staged W + WMMA f32)
    {
        dim3 grid((N_NODES / 16 + 7) / 8, D_OUT / WTILE);  // (782, 4)
        gcn_gemm_wmma<<<grid, 256, 0, stream>>>(x, weight, h);
    }

    // 2) out[r] += v * h[c]
    {
        gcn_spmm_scatter<<<(N_EDGES + 7) / 8, 256, 0, stream>>>(edge_row, edge_col,
                                                                edge_val, h, out);
    }
}